// DMTetGeometry_13443247637282
// MI455X (gfx1250) — compile-verified
//
#include <hip/hip_runtime.h>

// ---------------------------------------------------------------------------
// DMTet geometry extraction for MI455X (gfx1250, wave32).
// Memory-bound irregular pipeline: no GEMM content -> no WMMA (would be
// wasted); instead we lean on wave32 ballot compaction, LDS scans,
// L2-resident (192MB) hash dedup, CDNA5 async global->LDS loads, and the
// Tensor Data Mover (TDM) for bulk LDS staging in the streaming pass.
// ---------------------------------------------------------------------------

#define HSIZE_LOG 23
#define HSIZE (1u << HSIZE_LOG)          // 8.4M slots, 64MB keys -> L2-resident
#define ECAP  (1u << 22)                 // cap on unique crossing edges (4.2M)
#define BS    256

typedef __attribute__((ext_vector_type(4))) unsigned int tdm_u32x4;
typedef __attribute__((ext_vector_type(4))) int          tdm_i32x4;
typedef __attribute__((ext_vector_type(8))) int          tdm_i32x8;

// ---- marching tetrahedra tables (from reference) ----
__constant__ int c_TET_TABLE[16][12] = {
  {-1,-1,-1,-1,-1,-1,-1,-1,-1,-1,-1,-1},
  {0,4,5,6,-1,-1,-1,-1,-1,-1,-1,-1},
  {1,4,8,7,-1,-1,-1,-1,-1,-1,-1,-1},
  {7,1,8,6,5,1,7,6,5,0,1,6},
  {2,5,7,9,-1,-1,-1,-1,-1,-1,-1,-1},
  {4,0,6,7,9,0,7,6,7,0,9,2},
  {4,1,9,8,5,1,9,4,5,1,2,9},
  {6,0,1,2,8,6,1,2,9,6,8,2},
  {3,6,9,8,-1,-1,-1,-1,-1,-1,-1,-1},
  {5,0,4,8,5,0,8,3,5,8,9,3},
  {1,4,7,3,4,7,6,3,9,6,7,3},
  {0,1,5,3,5,1,9,3,5,1,7,9},
  {5,2,3,7,3,6,5,8,3,5,7,8},
  {0,4,7,8,0,3,8,7,0,3,7,2},
  {4,1,2,3,4,3,2,5,4,3,5,6},
  {0,1,2,3,-1,-1,-1,-1,-1,-1,-1,-1}};
__constant__ int c_NUM_TETS[16] = {0,1,1,3,1,3,3,3,1,3,3,3,3,3,3,1};

// ---------------------------------------------------------------------------
// K0: zero scratch (counters + hash keys + mark array) — must be re-zeroed
// every call since the harness does not re-poison between replays.
__global__ void zero_kernel(int* p, long long n) {
  long long i = (long long)blockIdx.x * BS + threadIdx.x;
  long long stride = (long long)gridDim.x * BS;
  for (; i < n; i += stride) p[i] = 0;
}

// ---------------------------------------------------------------------------
// K1: pos = verts + (1/res) * tanh(deform).  Streaming pass.
//  - deform tile: staged via TDM (tensor_load_to_lds builtin, clang-23
//    6-arg form) — one scalar-descriptor DMA per wave moves the wave's 128B
//    slice into the first BS floats of LDS, tracked on TENSORcnt. tensor_dim0 is clamped to
//    the remaining elements so OOB reads return zero; tile_dim0==0 on
//    fully-OOB tail waves => NOP per ISA. Groups 2/3 zero (<=2D tensor).
//  - verts: per-lane CDNA5 async global->LDS load into the second BS floats of LDS,
//    tracked on ASYNCcnt.
__global__ void pos_kernel(const float* __restrict__ verts,
                           const float* __restrict__ deform,
                           const int* __restrict__ gres,
                           float* __restrict__ pos, int n3) {
  __shared__ float lds[2 * BS];
  int tid = threadIdx.x;
  int i = blockIdx.x * BS + tid;

  // ---- TDM descriptor (D#) for this wave's deform tile ----
  unsigned wave = (unsigned)__builtin_amdgcn_readfirstlane(tid >> 5);  // wave32
  long long wbase = (long long)blockIdx.x * BS + (long long)wave * 32;
  int remain = n3 - (int)wbase;
  unsigned dim0 = remain > 0 ? (remain < 32 ? (unsigned)remain : 32u) : 0u;
  unsigned long long ga =
      (unsigned long long)(uintptr_t)(deform + (wbase < n3 ? wbase : 0));

#if __has_builtin(__builtin_amdgcn_tensor_load_to_lds)
  tdm_u32x4 g0;
  g0.x = 1u;                                               // count=1, user mode
  g0.y = wave * 128u;                                      // lds_addr (bytes)
  g0.z = (unsigned)ga;                                     // global_addr[95:64]
  g0.w = (unsigned)((ga >> 32) & 0x1ffffffu) | (2u << 30); // addr[120:96]|type=2
  tdm_i32x8 g1;
  g1.s0 = (int)(2u << 16);               // workgroup_mask=0, data_size=2 (4B)
  g1.s1 = (int)((dim0 & 0xffffu) << 16); // tensor_dim0[15:0]  (bits 63:48)
  g1.s2 = (int)(dim0 >> 16);             // tensor_dim0[31:16]; tensor_dim1=0
  g1.s3 = (int)((dim0 & 0xffffu) << 16); // tile_dim0 (bits 127:112); 0 => NOP
  g1.s4 = 0;                             // tile_dim1=0, tile_dim2=0
  g1.s5 = 32;                            // tensor_dim0_stride low (1D: unused)
  g1.s6 = 0;
  g1.s7 = 0;
  tdm_i32x4 z4 = {0, 0, 0, 0};
  tdm_i32x8 z8 = {0, 0, 0, 0, 0, 0, 0, 0};
  __builtin_amdgcn_tensor_load_to_lds(g0, g1, z4, z4, z8, 0);
  __builtin_amdgcn_s_wait_tensorcnt(0);
#else
  if (i < n3) lds[tid] = deform[i];      // fallback: plain staging
#endif

  // ---- per-lane async load of verts[i] into this lane's slot in the second half of LDS ----
  if (i < n3) {
    unsigned ldsoff = (unsigned)(BS * 4 + tid * 4);
    const float* gp = verts + i;
    asm volatile("global_load_async_to_lds_b32 %0, %1, off"
                 :: "v"(ldsoff), "v"(gp) : "memory");
  }
  asm volatile("s_wait_asynccnt 0" ::: "memory");

  if (i < n3) {
    float scale = 1.0f / (float)(*gres);      // 2/(2*res)
    pos[i] = lds[BS + tid] + scale * tanhf(lds[tid]);
  }
}

// ---------------------------------------------------------------------------
// K2: classify tets by occupancy; wave32 ballot-aggregated compaction into
// valid-tet and inner-tet lists (1 atomic per wave instead of per lane).
__global__ void classify_kernel(const int* __restrict__ indices,
                                const float* __restrict__ sdf, int F,
                                int* counters, int* vtets, int* tetidx,
                                int* itets) {
  int f = blockIdx.x * BS + threadIdx.x;
  bool valid = false, inner = false;
  int4 q = {0, 0, 0, 0};
  int ti = 0;
  if (f < F) {
    __builtin_prefetch(indices + 4 * (size_t)f + 512, 0, 0);  // global_prefetch_b8
    q = ((const int4*)indices)[f];
    int b0 = sdf[q.x] > 0.f, b1 = sdf[q.y] > 0.f;
    int b2 = sdf[q.z] > 0.f, b3 = sdf[q.w] > 0.f;
    ti = b0 | (b1 << 1) | (b2 << 2) | (b3 << 3);
    int s = b0 + b1 + b2 + b3;
    valid = (s > 0) && (s < 4);
    inner = (s == 4);
  }
  unsigned lane = threadIdx.x & 31u;           // wave32
  unsigned mv = (unsigned)__ballot(valid);
  int base = 0;
  if (lane == 0) base = mv ? atomicAdd(&counters[0], __popc(mv)) : 0;
  base = __shfl(base, 0, 32);
  if (valid) {
    int o = base + __popc(mv & ((1u << lane) - 1u));
    vtets[4 * o + 0] = q.x; vtets[4 * o + 1] = q.y;
    vtets[4 * o + 2] = q.z; vtets[4 * o + 3] = q.w;
    tetidx[o] = ti;
  }
  unsigned mi = (unsigned)__ballot(inner);
  if (lane == 0) base = mi ? atomicAdd(&counters[1], __popc(mi)) : 0;
  base = __shfl(base, 0, 32);
  if (inner) {
    int o = base + __popc(mi & ((1u << lane) - 1u));
    itets[4 * o + 0] = q.x; itets[4 * o + 1] = q.y;
    itets[4 * o + 2] = q.z; itets[4 * o + 3] = q.w;
  }
}

// ---------------------------------------------------------------------------
__device__ __forceinline__ unsigned hash64(unsigned long long k) {
  k ^= k >> 33; k *= 0xff51afd7ed558ccdULL;
  k ^= k >> 33; k *= 0xc4ceb9fe1a85ec53ULL;
  k ^= k >> 33;
  return (unsigned)k;
}
__device__ __constant__ int c_eA[6] = {0, 0, 0, 1, 1, 2};
__device__ __constant__ int c_eB[6] = {1, 2, 3, 2, 3, 3};

// K3: insert unique sign-crossing edges into open-addressing hash table.
// Key 0 is impossible (hi >= 1), so 0 == empty. Winner of the CAS allocates
// the edge id and records its endpoints; no spin-waits (lookup is a second
// kernel, so all hvals writes are visible by kernel boundary).
__global__ void edges_insert_kernel(int* counters, const int* __restrict__ vtets,
                                    const float* __restrict__ sdf,
                                    unsigned long long* hkeys, int* hvals,
                                    int* interp_v) {
  int i = blockIdx.x * BS + threadIdx.x;
  if (i >= counters[0]) return;
  int t[4] = {vtets[4 * i], vtets[4 * i + 1], vtets[4 * i + 2], vtets[4 * i + 3]};
#pragma unroll
  for (int e = 0; e < 6; ++e) {
    int a = t[c_eA[e]], b = t[c_eB[e]];
    int lo = min(a, b), hi = max(a, b);
    if ((sdf[lo] > 0.f) == (sdf[hi] > 0.f)) continue;   // not a crossing
    unsigned long long key =
        ((unsigned long long)(unsigned)lo << 32) | (unsigned)hi;
    unsigned h = hash64(key) & (HSIZE - 1u);
    for (;;) {
      unsigned long long prev = atomicCAS(&hkeys[h], 0ULL, key);
      if (prev == 0ULL) {
        int id = atomicAdd(&counters[2], 1);
        interp_v[2 * id] = lo; interp_v[2 * id + 1] = hi;
        hvals[h] = id;
        break;
      }
      if (prev == key) break;
      h = (h + 1) & (HSIZE - 1u);
    }
  }
}

// K4: second pass — resolve each tet-edge to its new-vertex id (-1 if none).
__global__ void edges_lookup_kernel(const int* counters, const int* __restrict__ vtets,
                                    const float* __restrict__ sdf,
                                    const unsigned long long* __restrict__ hkeys,
                                    const int* __restrict__ hvals,
                                    int* __restrict__ idxmap) {
  int i = blockIdx.x * BS + threadIdx.x;
  if (i >= counters[0]) return;
  int t[4] = {vtets[4 * i], vtets[4 * i + 1], vtets[4 * i + 2], vtets[4 * i + 3]};
#pragma unroll
  for (int e = 0; e < 6; ++e) {
    int a = t[c_eA[e]], b = t[c_eB[e]];
    int lo = min(a, b), hi = max(a, b);
    int id = -1;
    if ((sdf[lo] > 0.f) != (sdf[hi] > 0.f)) {
      unsigned long long key =
          ((unsigned long long)(unsigned)lo << 32) | (unsigned)hi;
      unsigned h = hash64(key) & (HSIZE - 1u);
      while (hkeys[h] != key) h = (h + 1) & (HSIZE - 1u);
      id = hvals[h];
    }
    idxmap[6 * i + e] = id;
  }
}

// ---------------------------------------------------------------------------
// K5: interpolate crossing-edge vertices: v = (p1*(-s2) + p2*s1)/(s1-s2)
__global__ void interp_kernel(const int* counters, const int* __restrict__ interp_v,
                              const float* __restrict__ sdf,
                              const float* __restrict__ pos,
                              float* __restrict__ nverts) {
  int i = blockIdx.x * BS + threadIdx.x;
  if (i >= counters[2]) return;
  int a = interp_v[2 * i], b = interp_v[2 * i + 1];
  float s1 = sdf[a], s2 = sdf[b];
  float inv = 1.0f / (s1 - s2);
  float w0 = -s2 * inv, w1 = s1 * inv;
#pragma unroll
  for (int c = 0; c < 3; ++c)
    nverts[3 * (size_t)i + c] = pos[3 * (size_t)a + c] * w0 + pos[3 * (size_t)b + c] * w1;
}

// ---------------------------------------------------------------------------
// K6: emit surface-adjacent tets via the marching-tet table; mark used verts.
__global__ void emit_side_kernel(int* counters, const int* __restrict__ vtets,
                                 const int* __restrict__ tetidx,
                                 const int* __restrict__ idxmap, int N,
                                 int* __restrict__ alltets, int* __restrict__ mark) {
  int i = blockIdx.x * BS + threadIdx.x;
  if (i >= counters[0]) return;
  int ti = tetidx[i];
  int nt = c_NUM_TETS[ti];
  int tve[10];
#pragma unroll
  for (int k = 0; k < 4; ++k) tve[k] = vtets[4 * i + k];
#pragma unroll
  for (int k = 0; k < 6; ++k) tve[4 + k] = idxmap[6 * i + k] + N;
  int base = atomicAdd(&counters[3], nt);
  for (int k = 0; k < nt; ++k)
#pragma unroll
    for (int c = 0; c < 4; ++c) {
      int v = tve[c_TET_TABLE[ti][4 * k + c]];
      alltets[4 * (size_t)(base + k) + c] = v;
      mark[v] = 1;
    }
}

// K7: append fully-interior tets after the side tets.
__global__ void emit_inner_kernel(const int* counters, const int* __restrict__ itets,
                                  int* __restrict__ alltets, int* __restrict__ mark) {
  int j = blockIdx.x * BS + threadIdx.x;
  if (j >= counters[1]) return;
  int base = counters[3];
#pragma unroll
  for (int c = 0; c < 4; ++c) {
    int v = itets[4 * (size_t)j + c];
    alltets[4 * (size_t)(base + j) + c] = v;
    mark[v] = 1;
  }
}

// ---------------------------------------------------------------------------
// K8-K10: exclusive scan of mark[0..N+ECAP) — reproduces jnp.unique's sorted
// ascending renumbering exactly. LDS Hillis-Steele per block + serial block-
// sum scan + add-offsets/gather pass.
__global__ void scan_block_kernel(const int* __restrict__ mark, int* __restrict__ newid,
                                  int* __restrict__ bsums, int M) {
  __shared__ int s[BS];
  int tid = threadIdx.x;
  int i = blockIdx.x * BS + tid;
  int v = (i < M) ? mark[i] : 0;
  s[tid] = v;
  __syncthreads();
#pragma unroll
  for (int off = 1; off < BS; off <<= 1) {
    int t = (tid >= off) ? s[tid - off] : 0;
    __syncthreads();
    s[tid] += t;
    __syncthreads();
  }
  if (i < M) newid[i] = s[tid] - v;        // exclusive prefix within block
  if (tid == BS - 1) bsums[blockIdx.x] = s[BS - 1];
}

__global__ void scan_sums_kernel(int* bsums, int NB) {
  __shared__ int s[BS];
  __shared__ int carry;
  int tid = threadIdx.x;
  if (tid == 0) carry = 0;
  __syncthreads();
  for (int base = 0; base < NB; base += BS) {
    int i = base + tid;
    int v = (i < NB) ? bsums[i] : 0;
    s[tid] = v;
    __syncthreads();
#pragma unroll
    for (int off = 1; off < BS; off <<= 1) {
      int t = (tid >= off) ? s[tid - off] : 0;
      __syncthreads();
      s[tid] += t;
      __syncthreads();
    }
    int excl = s[tid] - v + carry;
    if (i < NB) bsums[i] = excl;
    int tot = s[BS - 1];
    __syncthreads();
    if (tid == 0) carry += tot;
    __syncthreads();
  }
}

__global__ void scan_add_kernel(const int* __restrict__ mark, int* __restrict__ newid,
                                const int* __restrict__ bsums, int M, int N,
                                const float* __restrict__ pos,
                                const float* __restrict__ nverts,
                                float* __restrict__ outf, int* counters) {
  int i = blockIdx.x * BS + threadIdx.x;
  if (i >= M) return;
  int id = newid[i] + bsums[blockIdx.x];
  newid[i] = id;
  if (mark[i]) {
    const float* src = (i < N) ? (pos + 3 * (size_t)i) : (nverts + 3 * (size_t)(i - N));
    outf[3 * (size_t)id + 0] = src[0];
    outf[3 * (size_t)id + 1] = src[1];
    outf[3 * (size_t)id + 2] = src[2];
  }
  if (i == M - 1) counters[4] = id + mark[i];   // total unique verts
}

// ---------------------------------------------------------------------------
// K11: remap tets through newid into the tail of d_out (int32 bit patterns).
__global__ void remap_kernel(const int* counters, const int* __restrict__ alltets,
                             const int* __restrict__ newid, int* __restrict__ outi) {
  long long k = (long long)blockIdx.x * BS + threadIdx.x;
  long long nT = (long long)counters[3] + counters[1];
  if (k >= 4 * nT) return;
  long long base = 3LL * counters[4];
  outi[base + k] = newid[alltets[k]];
}

// ---------------------------------------------------------------------------
extern "C" void kernel_launch(void* const* d_in, const int* in_sizes, int n_in,
                              void* d_out, int out_size, void* d_ws, size_t ws_size,
                              hipStream_t stream) {
  (void)n_in; (void)out_size; (void)ws_size;
  const float* verts  = (const float*)d_in[0];
  const float* deform = (const float*)d_in[1];
  const float* sdf    = (const float*)d_in[2];
  const int*   idx    = (const int*)d_in[3];
  const int*   gres   = (const int*)d_in[4];

  const int N3 = in_sizes[0];
  const int N  = in_sizes[2];
  const int F  = in_sizes[3] / 4;
  const int M  = N + (int)ECAP;          // mark/newid domain
  const int NB = (M + BS - 1) / BS;

  // ---- workspace layout (256B aligned regions) ----
  char* ws = (char*)d_ws;
  size_t off = 0;
  auto alloc = [&](size_t bytes) -> void* {
    void* p = ws + off;
    off = (off + bytes + 255) & ~(size_t)255;
    return p;
  };
  // zero-initialized span first:
  int*                 counters = (int*)alloc(64 * sizeof(int));
  unsigned long long*  hkeys    = (unsigned long long*)alloc((size_t)HSIZE * 8);
  int*                 mark     = (int*)alloc((size_t)M * 4);
  size_t zero_ints = off / 4;
  // overwrite-only regions:
  int*   hvals   = (int*)alloc((size_t)HSIZE * 4);
  float* pos     = (float*)alloc((size_t)N3 * 4);
  int*   vtets   = (int*)alloc((size_t)F * 4 * 4);
  int*   tetidx  = (int*)alloc((size_t)F * 4);
  int*   itets   = (int*)alloc((size_t)F * 4 * 4);
  int*   idxmap  = (int*)alloc((size_t)F * 6 * 4);
  int*   interp  = (int*)alloc((size_t)ECAP * 2 * 4);
  float* nverts  = (float*)alloc((size_t)ECAP * 3 * 4);
  int*   alltets = (int*)alloc((size_t)F * 4 * 4 * 4);   // up to 4 tets/input tet
  int*   newid   = (int*)alloc((size_t)M * 4);
  int*   bsums   = (int*)alloc((size_t)NB * 4);

  dim3 blk(BS);
  int gF  = (F + BS - 1) / BS;
  int gN3 = (N3 + BS - 1) / BS;
  int gE  = ((int)ECAP + BS - 1) / BS;
  long long totTet = 16LL * F;
  int gT  = (int)((totTet + BS - 1) / BS);

  zero_kernel<<<4096, blk, 0, stream>>>((int*)d_ws, (long long)zero_ints);
  pos_kernel<<<gN3, blk, 0, stream>>>(verts, deform, gres, pos, N3);
  classify_kernel<<<gF, blk, 0, stream>>>(idx, sdf, F, counters, vtets, tetidx, itets);
  edges_insert_kernel<<<gF, blk, 0, stream>>>(counters, vtets, sdf, hkeys, hvals, interp);
  edges_lookup_kernel<<<gF, blk, 0, stream>>>(counters, vtets, sdf, hkeys, hvals, idxmap);
  interp_kernel<<<gE, blk, 0, stream>>>(counters, interp, sdf, pos, nverts);
  emit_side_kernel<<<gF, blk, 0, stream>>>(counters, vtets, tetidx, idxmap, N, alltets, mark);
  emit_inner_kernel<<<gF, blk, 0, stream>>>(counters, itets, alltets, mark);
  scan_block_kernel<<<NB, blk, 0, stream>>>(mark, newid, bsums, M);
  scan_sums_kernel<<<1, blk, 0, stream>>>(bsums, NB);
  scan_add_kernel<<<NB, blk, 0, stream>>>(mark, newid, bsums, M, N, pos, nverts,
                                          (float*)d_out, counters);
  remap_kernel<<<gT, blk, 0, stream>>>(counters, alltets, newid, (int*)d_out);
}